// TripletLoss_80427557585292
// MI455X (gfx1250) — compile-verified
//
#include <hip/hip_runtime.h>

typedef __attribute__((ext_vector_type(2))) float v2f;
typedef __attribute__((ext_vector_type(8))) float v8f;

#define DIM 128

// ---------------------------------------------------------------------------
// Kernel 1: row norms  nrm[r] = sum_d x[r][d]^2   (one wave32 per row)
// ---------------------------------------------------------------------------
__global__ void tl_norms_kernel(const float* __restrict__ x,
                                float* __restrict__ nrm, int N) {
  const int lane = threadIdx.x & 31;
  const int row  = blockIdx.x * (blockDim.x >> 5) + (threadIdx.x >> 5);
  if (row >= N) return;
  const float4 v = *(const float4*)(x + (size_t)row * DIM + lane * 4);
  float s = v.x * v.x + v.y * v.y + v.z * v.z + v.w * v.w;
  #pragma unroll
  for (int off = 16; off; off >>= 1) s += __shfl_xor(s, off, 32);
  if (lane == 0) nrm[row] = s;
}

// ---------------------------------------------------------------------------
// Kernel 2: one wave32 per tile of 16 triplets.
//   A (16x4 f32)  = rows    x[i_t]   -> lane m holds K, K+1 (K+=2 for hi half)
//   B (4x16 f32)  = columns x[j_t]   -> same per-lane layout, N in lanes
//   accJ/accK accumulate over K=0..127 via V_WMMA_F32_16X16X4_F32.
//   Diagonal (t,t) of the 16x16 accumulator = xi.xj for triplet t.
// ---------------------------------------------------------------------------
__global__ void tl_triplet_wmma_kernel(const float* __restrict__ x,
                                       const long long* __restrict__ trip,
                                       const float* __restrict__ nrm,
                                       float* __restrict__ partial,
                                       int T, int nTiles) {
  const int lane = threadIdx.x & 31;
  const int tile = blockIdx.x * (blockDim.x >> 5) + (threadIdx.x >> 5);
  if (tile >= nTiles) return;              // whole-wave branch: EXEC stays all-1s

  const int m  = lane & 15;                // M for A-frag, N for B-frag, triplet slot
  const int t  = tile * 16 + m;            // triplet handled by this lane slot
  const int tc = (t < T) ? t : (T - 1);    // clamp for ragged tail (masked later)

  const long long i = trip[3ll * tc + 0];
  const long long j = trip[3ll * tc + 1];
  const long long k = trip[3ll * tc + 2];

  const float* xi = x + (size_t)i * DIM;
  const float* xj = x + (size_t)j * DIM;
  const float* xk = x + (size_t)k * DIM;
  const int koff = (lane < 16) ? 0 : 2;    // hi lane-half covers K+2,K+3

  v8f accJ = {};
  v8f accK = {};
  #pragma unroll 8
  for (int kb = 0; kb < DIM; kb += 4) {
    v2f a  = *(const v2f*)(xi + kb + koff);
    v2f bj = *(const v2f*)(xj + kb + koff);
    v2f bk = *(const v2f*)(xk + kb + koff);
    accJ = __builtin_amdgcn_wmma_f32_16x16x4_f32(false, a, false, bj,
                                                 (short)0, accJ, false, false);
    accK = __builtin_amdgcn_wmma_f32_16x16x4_f32(false, a, false, bk,
                                                 (short)0, accK, false, false);
  }

  // Diagonal extraction: element (t,t) sits in lane t (t<8, VGPR t) or
  // lane t+16 (t>=8, VGPR t-8).  Each diag lane's own (i,j,k) == triplet t.
  float sp = 0.0f;
  const bool diagLo = (lane < 8);
  const bool diagHi = (lane >= 24);
  if (diagLo || diagHi) {
    const int r = diagLo ? lane : (lane - 24);
    float dJ = 0.0f, dK = 0.0f;
    #pragma unroll
    for (int q = 0; q < 8; ++q) {
      if (r == q) { dJ = accJ[q]; dK = accK[q]; }
    }
    const float ni = nrm[i], nj = nrm[j], nk = nrm[k];
    const float dij  = fmaxf(ni + nj - 2.0f * dJ, 0.0f);
    const float dik  = fmaxf(ni + nk - 2.0f * dK, 0.0f);
    const float diff = dij - dik;
    // softplus(diff) = max(diff,0) + log1p(exp(-|diff|))   (numerically stable)
    sp = fmaxf(diff, 0.0f) + log1pf(expf(-fabsf(diff)));
    if (t >= T) sp = 0.0f;                 // mask ragged tail
  }

  #pragma unroll
  for (int off = 16; off; off >>= 1) sp += __shfl_xor(sp, off, 32);
  if (lane == 0) partial[tile] = sp;       // deterministic per-tile partial
}

// ---------------------------------------------------------------------------
// Kernel 3: fixed-order reduction of tile partials, scale by 1/T.
// ---------------------------------------------------------------------------
__global__ void tl_reduce_kernel(const float* __restrict__ partial, int nTiles,
                                 float invT, float* __restrict__ out) {
  __shared__ float sm[256];
  float s = 0.0f;
  for (int ti = threadIdx.x; ti < nTiles; ti += blockDim.x) s += partial[ti];
  sm[threadIdx.x] = s;
  __syncthreads();
  for (int w = blockDim.x >> 1; w; w >>= 1) {
    if ((int)threadIdx.x < w) sm[threadIdx.x] += sm[threadIdx.x + w];
    __syncthreads();
  }
  if (threadIdx.x == 0) out[0] = sm[0] * invT;
}

// ---------------------------------------------------------------------------
extern "C" void kernel_launch(void* const* d_in, const int* in_sizes, int n_in,
                              void* d_out, int out_size, void* d_ws, size_t ws_size,
                              hipStream_t stream) {
  const float*     x    = (const float*)d_in[0];
  const long long* trip = (const long long*)d_in[1];   // int64 triplets (T,3)

  const int N      = in_sizes[0] / DIM;   // 8192
  const int T      = in_sizes[1] / 3;     // 200000
  const int nTiles = (T + 15) / 16;       // 12500

  float* nrm     = (float*)d_ws;          // N floats
  float* partial = nrm + N;               // nTiles floats

  const int threads       = 256;          // 8 wave32s per block
  const int wavesPerBlock = threads >> 5;

  tl_norms_kernel<<<(N + wavesPerBlock - 1) / wavesPerBlock, threads, 0, stream>>>(
      x, nrm, N);

  tl_triplet_wmma_kernel<<<(nTiles + wavesPerBlock - 1) / wavesPerBlock, threads,
                           0, stream>>>(x, trip, nrm, partial, T, nTiles);

  tl_reduce_kernel<<<1, threads, 0, stream>>>(partial, nTiles, 1.0f / (float)T,
                                              (float*)d_out);
}